// LambdaRankLoss_41515153883612
// MI455X (gfx1250) — compile-verified
//
#include <hip/hip_runtime.h>
#include <hip/hip_bf16.h>

#define N       256
#define ROWS    8            // batch rows per block
#define THREADS 256
#define WAVES   (THREADS / 32)
#define EPSV    1e-6f
#define LOG2E   1.44269504088896f
#define LN2     0.69314718055994531f
#define MASK_WORDS (THREADS * 4)   // 256 threads x 128 bits = 1024 u32

typedef float v2f __attribute__((ext_vector_type(2)));
typedef float v8f __attribute__((ext_vector_type(8)));

// ---------------------------------------------------------------------------
// Zero the cross-block validity bitmask (must happen every call: graph replay
// does not re-poison workspace, and we OR into it).
// ---------------------------------------------------------------------------
__global__ void lrank_init(unsigned* __restrict__ g_mask) {
    int i = blockIdx.x * blockDim.x + threadIdx.x;
    if (i < MASK_WORDS) g_mask[i] = 0u;
}

// ---------------------------------------------------------------------------
// Main kernel: each block processes ROWS batch rows over all 32640 pairs.
// Folded triangular iteration: iteration ii in [0,128) covers row ii
// (upper fold) and row 255-ii (lower fold) => 255 active lanes / iteration.
// Per-wave f32 reduction uses V_WMMA_F32_16X16X4_F32 with a ones B-matrix.
// ---------------------------------------------------------------------------
__global__ __launch_bounds__(THREADS) void lrank_main(
    const float* __restrict__ pred, const float* __restrict__ tru,
    unsigned* __restrict__ g_mask, float* __restrict__ g_partial)
{
    __shared__ float sp[ROWS * N];
    __shared__ float st[ROWS * N];
    __shared__ float s_wsum[WAVES];

    const int  t    = threadIdx.x;
    const long base = (long)blockIdx.x * (ROWS * N);

    // Coalesced stage of ROWS rows of both tensors into LDS (16 KB).
    for (int k = t; k < ROWS * N; k += THREADS) {
        sp[k] = pred[base + k];
        st[k] = tru [base + k];
    }
    __syncthreads();

    float    acc0 = 0.f, acc1 = 0.f;        // 2 accumulators -> 16x4 A matrix
    unsigned vmask[4] = {0u, 0u, 0u, 0u};   // per-thread any-valid bits over ii
    const bool live = (t != N - 1);         // lane 255 is the one idle slot

    for (int ii = 0; ii < N / 2; ++ii) {
        const bool inA = (t < (N - 1 - ii));
        const int  row = inA ? ii : (N - 1 - ii);
        int        j   = inA ? (ii + 1 + t) : (t + 1);
        if (j > N - 1) j = N - 1;           // clamp for dead lane 255

        unsigned anyv = 0u;
        float    accL = 0.f;
#pragma unroll
        for (int rr = 0; rr < ROWS; ++rr) {
            const int ro = rr * N;
            const float ti = st[ro + row], tj = st[ro + j];
            const float pi = sp[ro + row], pj = sp[ro + j];
            const float dt  = ti - tj;
            const float dp  = pi - pj;
            const float adt = fabsf(dt);
            const bool  valid = (adt >= EPSV) && live;
            const float s = (dt >= 0.f) ? dp : -dp;     // sign(dt)*dp
            const float x = -s;
            // stable softplus: log1p(exp(x)) = max(x,0) + log1p(exp(-|x|))
            // hardware transcendentals: v_exp_f32 (2^x) and v_log_f32 (log2 x)
            const float e  = __builtin_amdgcn_exp2f(-fabsf(x) * LOG2E);
            const float pl = fmaxf(x, 0.f) + LN2 * __builtin_amdgcn_logf(1.f + e);
            accL += valid ? pl * adt : 0.f;
            anyv |= valid ? 1u : 0u;
        }
        if (ii & 1) acc1 += accL; else acc0 += accL;
        vmask[ii >> 5] |= anyv << (ii & 31);
    }

    // Publish validity bits (OR is idempotent & commutative => deterministic).
#pragma unroll
    for (int w = 0; w < 4; ++w)
        atomicOr(&g_mask[t * 4 + w], vmask[w]);

    // ---- wave reduction via V_WMMA_F32_16X16X4_F32 ----
    // A (16x4 f32): VGPR0 = acc0, VGPR1 = acc1  -> 64 f32 values per wave.
    // B = all-ones 4x16  ->  D[m][n] = rowsum(m) for every n.
    v2f a; a.x = acc0; a.y = acc1;
    v2f b; b.x = 1.0f; b.y = 1.0f;
    v8f c = {};
    c = __builtin_amdgcn_wmma_f32_16x16x4_f32(
            /*neg_a=*/false, a, /*neg_b=*/false, b,
            /*c_mod=*/(short)0, c, /*reuse_a=*/false, /*reuse_b=*/false);
    // lanes 0-15 hold rowsums m=0..7 in c[0..7]; lanes 16-31 hold m=8..15.
    float h = c[0] + c[1] + c[2] + c[3] + c[4] + c[5] + c[6] + c[7];
    float wave_total = h + __shfl_xor(h, 16, 32);   // total of all 64 values

    const int lane = t & 31, wv = t >> 5;
    if (lane == 0) s_wsum[wv] = wave_total;
    __syncthreads();
    if (t == 0) {
        float bsum = 0.f;
#pragma unroll
        for (int w = 0; w < WAVES; ++w) bsum += s_wsum[w];
        g_partial[blockIdx.x] = bsum;
    }
}

// ---------------------------------------------------------------------------
// Final: popcount validity mask, deterministically sum block partials,
// emit scalar = (grand_sum / B) / count   (or grand_sum / B if count == 0).
// ---------------------------------------------------------------------------
__global__ __launch_bounds__(THREADS) void lrank_final(
    const unsigned* __restrict__ g_mask, const float* __restrict__ g_partial,
    int nblocks, int Bsz, float* __restrict__ out)
{
    __shared__ float ssum[THREADS];
    __shared__ int   scnt[THREADS];
    const int t = threadIdx.x;

    float s = 0.f;
    for (int i = t; i < nblocks; i += THREADS) s += g_partial[i];

    int cnt = 0;
#pragma unroll
    for (int w = 0; w < 4; ++w) cnt += __builtin_popcount(g_mask[t * 4 + w]);

    ssum[t] = s; scnt[t] = cnt;
    __syncthreads();
    for (int o = THREADS / 2; o > 0; o >>= 1) {
        if (t < o) { ssum[t] += ssum[t + o]; scnt[t] += scnt[t + o]; }
        __syncthreads();
    }
    if (t == 0) {
        const float meanB = ssum[0] / (float)Bsz;
        const int   c     = scnt[0];
        out[0] = (c > 0) ? (meanB / (float)c) : meanB;
    }
}

// ---------------------------------------------------------------------------
extern "C" void kernel_launch(void* const* d_in, const int* in_sizes, int n_in,
                              void* d_out, int out_size, void* d_ws, size_t ws_size,
                              hipStream_t stream) {
    (void)n_in; (void)out_size; (void)ws_size;
    const float* pred = (const float*)d_in[0];
    const float* tru  = (const float*)d_in[1];
    float*       out  = (float*)d_out;

    const int Bsz     = in_sizes[0] / N;   // 4096
    const int nblocks = Bsz / ROWS;        // 512

    unsigned* g_mask    = (unsigned*)d_ws;                                 // 4 KB
    float*    g_partial = (float*)((char*)d_ws + MASK_WORDS * sizeof(unsigned));

    lrank_init <<<(MASK_WORDS + 255) / 256, 256, 0, stream>>>(g_mask);
    lrank_main <<<nblocks, THREADS, 0, stream>>>(pred, tru, g_mask, g_partial);
    lrank_final<<<1, THREADS, 0, stream>>>(g_mask, g_partial, nblocks, Bsz, out);
}